// LSTMModel_46127948759467
// MI455X (gfx1250) — compile-verified
//
#include <hip/hip_runtime.h>

// LSTM: B=4096, T=512, I=8, H=64, O=1.  fp32 end-to-end via V_WMMA_F32_16X16X4_F32.
//
// Block = 128 threads (4 waves), owns 16 batch rows.
// Wave w owns hidden cols j in [16w, 16w+16) and gate n-tiles {w, w+4, w+8, w+12}
// -> i/f/g/o for a given (row,j) land in the same lane & same acc register index,
//    so the cell update is lane-local VALU, no cross-lane traffic.
// h is exchanged between the 4 waves via a ping-pong LDS buffer (1 barrier/step).
// W_hh / W_ih B-tiles are loop-invariant and pinned in VGPRs.
// Activations use hardware V_TANH_F32 when the builtin exists (sigmoid via
// 0.5*tanh(x/2)+0.5), halving the transcendental chain on the serial path.

typedef __attribute__((ext_vector_type(2))) float v2f;
typedef __attribute__((ext_vector_type(8))) float v8f;

#define NB   4096   // batch
#define NT   512    // timesteps
#define NI   8      // input size
#define NH   64     // hidden size
#define HS   68     // padded LDS row stride (floats): (row*68 + col) % 64 spreads
                    // the A-tile b64 pattern across all 64 banks conflict-free.

// ---- hardware tanh if available (CDNA5 lists TANH as a TRANS VALU op) ----
#if defined(__has_builtin)
#  if __has_builtin(__builtin_amdgcn_tanhf)
#    define HAVE_HW_TANH 1
__device__ __forceinline__ float hw_tanh(float x) { return __builtin_amdgcn_tanhf(x); }
#  elif __has_builtin(__builtin_amdgcn_tanh_f32)
#    define HAVE_HW_TANH 1
__device__ __forceinline__ float hw_tanh(float x) { return __builtin_amdgcn_tanh_f32(x); }
#  endif
#endif
#ifndef HAVE_HW_TANH
#  define HAVE_HW_TANH 0
// fallback: 1 - 2/(e^{2x}+1); saturates correctly at +-1
__device__ __forceinline__ float hw_tanh(float x) {
    return 1.0f - 2.0f * __builtin_amdgcn_rcpf(__expf(x + x) + 1.0f);
}
#endif

__device__ __forceinline__ float fast_sigmoid(float x) {
#if HAVE_HW_TANH
    return __builtin_fmaf(0.5f, hw_tanh(0.5f * x), 0.5f);   // 1 trans + 2 valu
#else
    return __builtin_amdgcn_rcpf(1.0f + __expf(-x));
#endif
}

__global__ __launch_bounds__(128) void lstm_wmma_f32_kernel(
    const float* __restrict__ x,     // [NB, NT, NI]
    const float* __restrict__ W_ih,  // [4H, NI]
    const float* __restrict__ W_hh,  // [4H, NH]
    const float* __restrict__ b_ih,  // [4H]
    const float* __restrict__ b_hh,  // [4H]
    const float* __restrict__ W_fc,  // [1, NH]
    const float* __restrict__ b_fc,  // [1]
    float* __restrict__ out)         // [NB, 1]
{
    __shared__ float hbuf[2][16 * HS];   // ping-pong h exchange, 16 rows x 64(+pad)

    const int tid  = threadIdx.x;
    const int wv   = tid >> 5;          // wave 0..3 -> hidden cols [16wv, 16wv+16)
    const int lane = tid & 31;
    const int lm   = lane & 15;         // N / M position within a 16-wide tile
    const int hi   = lane >> 4;         // half-wave select (K pairing for f32 tiles)
    const int row0 = blockIdx.x * 16;   // batch rows owned by this block

    // ---- Pin loop-invariant B-matrix tiles (W_hh, W_ih) + fused bias in VGPRs ----
    // B-tile (4x16 f32) per lane: 2 consecutive K at N = nrow:  {K=4kc+2hi, +1}
    v2f  Bhh[4][16];   // gt = gate (i,f,g,o), kc = K-chunk of 4 over NH=64
    v2f  Bih[4][2];    // K-chunks over NI=8
    float biasv[4];
    #pragma unroll
    for (int gt = 0; gt < 4; ++gt) {
        const int nrow = (4 * gt + wv) * 16 + lm;   // gate row in [0,256)
        #pragma unroll
        for (int kc = 0; kc < 16; ++kc)
            Bhh[gt][kc] = *(const v2f*)(W_hh + nrow * NH + 4 * kc + 2 * hi);
        #pragma unroll
        for (int kc = 0; kc < 2; ++kc)
            Bih[gt][kc] = *(const v2f*)(W_ih + nrow * NI + 4 * kc + 2 * hi);
        biasv[gt] = b_ih[nrow] + b_hh[nrow];
    }

    // h0 = 0
    for (int idx = tid; idx < 16 * HS; idx += 128) hbuf[0][idx] = 0.0f;
    __syncthreads();

    v8f c = {0.f, 0.f, 0.f, 0.f, 0.f, 0.f, 0.f, 0.f};   // cell state, 16x16 C-layout

    #pragma unroll 1
    for (int t = 0; t < NT; ++t) {
        const int p = t & 1;
        const float* __restrict__ hsrc = hbuf[p];

        // A tiles of h (16 rows x K=64): one ds_load_b64 per K-chunk per lane
        v2f Ah[16];
        #pragma unroll
        for (int kc = 0; kc < 16; ++kc)
            Ah[kc] = *(const v2f*)(hsrc + lm * HS + 4 * kc + 2 * hi);

        // A tiles of x_t (16 rows x K=8), straight from global (64MB streamed total)
        v2f Ax[2];
        #pragma unroll
        for (int kc = 0; kc < 2; ++kc)
            Ax[kc] = *(const v2f*)(x + ((row0 + lm) * NT + t) * NI + 4 * kc + 2 * hi);

        // acc = bias (bias depends only on N -> identical across C rows/registers)
        v8f acc[4];
        #pragma unroll
        for (int gt = 0; gt < 4; ++gt)
            #pragma unroll
            for (int r = 0; r < 8; ++r) acc[gt][r] = biasv[gt];

        // gates += x_t @ W_ih^T   (2 WMMAs per gate tile)
        #pragma unroll
        for (int gt = 0; gt < 4; ++gt) {
            acc[gt] = __builtin_amdgcn_wmma_f32_16x16x4_f32(
                false, Ax[0], false, Bih[gt][0], (short)0, acc[gt], false, false);
            acc[gt] = __builtin_amdgcn_wmma_f32_16x16x4_f32(
                false, Ax[1], false, Bih[gt][1], (short)0, acc[gt], false, false);
        }
        // gates += h @ W_hh^T   (16 K-chunks x 4 gate tiles = 64 WMMAs,
        //  inner gt loop gives 4 independent accumulation chains for the XDL pipe)
        #pragma unroll
        for (int kc = 0; kc < 16; ++kc)
            #pragma unroll
            for (int gt = 0; gt < 4; ++gt)
                acc[gt] = __builtin_amdgcn_wmma_f32_16x16x4_f32(
                    false, Ah[kc], false, Bhh[gt][kc], (short)0, acc[gt], false, false);

        // Lane-local cell update: acc[0]=i, acc[1]=f, acc[2]=g, acc[3]=o,
        // all aligned per (row, j) by construction.
        float* __restrict__ hdst = hbuf[p ^ 1];
        #pragma unroll
        for (int r = 0; r < 8; ++r) {
            const float iv = fast_sigmoid(acc[0][r]);
            const float fv = fast_sigmoid(acc[1][r]);
            const float gv = hw_tanh     (acc[2][r]);
            const float ov = fast_sigmoid(acc[3][r]);
            const float cn = fv * c[r] + iv * gv;
            c[r] = cn;
            hdst[(r + 8 * hi) * HS + 16 * wv + lm] = ov * hw_tanh(cn);
        }
        __syncthreads();   // publishes hbuf[p^1]; hbuf[p] reads finished above
    }

    // Final projection: h_T sits in hbuf[NT & 1] == hbuf[0].  O=1, trivial.
    if (tid < 16) {
        float accv = b_fc[0];
        const float* hr = hbuf[0] + tid * HS;
        #pragma unroll
        for (int j = 0; j < NH; ++j) accv += hr[j] * W_fc[j];
        out[row0 + tid] = accv;
    }
}

extern "C" void kernel_launch(void* const* d_in, const int* in_sizes, int n_in,
                              void* d_out, int out_size, void* d_ws, size_t ws_size,
                              hipStream_t stream) {
    const float* x    = (const float*)d_in[0];
    const float* W_ih = (const float*)d_in[1];
    const float* W_hh = (const float*)d_in[2];
    const float* b_ih = (const float*)d_in[3];
    const float* b_hh = (const float*)d_in[4];
    const float* W_fc = (const float*)d_in[5];
    const float* b_fc = (const float*)d_in[6];
    float* out = (float*)d_out;

    lstm_wmma_f32_kernel<<<NB / 16, 128, 0, stream>>>(
        x, W_ih, W_hh, b_ih, b_hh, W_fc, b_fc, out);
}